// Sampler_62929860821592
// MI455X (gfx1250) — compile-verified
//
#include <hip/hip_runtime.h>
#include <hip/hip_bf16.h>
#include <math.h>
#include <limits.h>

#define BD 1024  // threads per block = 32 waves (wave32)

// ---- gfx1250 async-to-LDS path (probe via __has_builtin; fallback = plain loads)
#if defined(__AMDGCN__) && __has_builtin(__builtin_amdgcn_global_load_async_to_lds_b128)
#define ASYNC_PIPE 1
#else
#define ASYNC_PIPE 0
#endif

#if ASYNC_PIPE
#if __has_builtin(__builtin_amdgcn_s_wait_asynccnt)
#define WAIT_ASYNC(n) __builtin_amdgcn_s_wait_asynccnt(n)
#else
#define WAIT_ASYNC(n) asm volatile("s_wait_asynccnt %0" ::"i"(n) : "memory")
#endif
typedef int v4i_t __attribute__((ext_vector_type(4)));
typedef __attribute__((address_space(1))) v4i_t gas_v4i;  // global
typedef __attribute__((address_space(3))) v4i_t las_v4i;  // LDS
__device__ __forceinline__ void async_copy16(const float4* gsrc, float4* ldst) {
  __builtin_amdgcn_global_load_async_to_lds_b128(
      (gas_v4i*)gsrc, (las_v4i*)ldst, /*offset=*/0, /*cpol=*/0);
}
#endif

// Order-preserving float -> uint mapping (descending float == descending uint)
__device__ __forceinline__ unsigned fkey(float f) {
  unsigned b = __float_as_uint(f);
  unsigned m = (unsigned)((int)b >> 31) | 0x80000000u;
  return b ^ m;
}

__device__ __forceinline__ unsigned rotl32(unsigned v, int n) {
  return (v << n) | (v >> (32 - n));
}

// JAX threefry2x32, key = (0, 42) (jax.random.key(42)), 20 rounds
__device__ __forceinline__ void threefry2x32_42(unsigned c0, unsigned c1,
                                                unsigned& o0, unsigned& o1) {
  const unsigned k0 = 0u, k1 = 42u;
  unsigned ks[3] = {k0, k1, k0 ^ k1 ^ 0x1BD11BDAu};
  unsigned x0 = c0 + ks[0], x1 = c1 + ks[1];
  const int rot[2][4] = {{13, 15, 26, 6}, {17, 29, 16, 24}};
#pragma unroll
  for (int i = 0; i < 5; ++i) {
    const int* r = rot[i & 1];
#pragma unroll
    for (int j = 0; j < 4; ++j) {
      x0 += x1;
      x1 = rotl32(x1, r[j]);
      x1 ^= x0;
    }
    x0 += ks[(i + 1) % 3];
    x1 += ks[(i + 2) % 3] + (unsigned)(i + 1);
  }
  o0 = x0;
  o1 = x1;
}

// Gumbel noise for flat element index, matching JAX's counter layout:
// counts = iota(n); pairs (i, i+n/2); out[0:n/2]=x0, out[n/2:n]=x1
__device__ __forceinline__ float gumbel_noise(unsigned flat, unsigned halfN) {
  unsigned c0 = (flat < halfN) ? flat : (flat - halfN);
  unsigned o0, o1;
  threefry2x32_42(c0, c0 + halfN, o0, o1);
  unsigned bits = (flat < halfN) ? o0 : o1;
  float f = __uint_as_float((bits >> 9) | 0x3F800000u) - 1.0f;  // [0,1)
  const float TINY = 1.17549435e-38f;
  float u = __fadd_rn(__fmul_rn(f, 1.0f - TINY), TINY);  // uniform(minval=tiny)
  u = fmaxf(u, TINY);
  return -logf(-logf(u));
}

// Find, among NB = BD*G histogram bins scanned in DESCENDING bin order, the bin
// where the cumulative count first reaches k. Publishes bin index and the
// residual rank inside that bin.  Wave32 two-level inclusive scan.
template <int G>
__device__ __forceinline__ void selectStep(const unsigned* hist, unsigned NB, unsigned k,
                                           unsigned* s_warp, unsigned* s_selBin,
                                           unsigned* s_remain) {
  const unsigned tid = threadIdx.x;
  const unsigned lane = tid & 31u, wave = tid >> 5;
  unsigned v[G];
  unsigned s = 0;
#pragma unroll
  for (int j = 0; j < G; ++j) {
    v[j] = hist[NB - 1u - tid * G - (unsigned)j];
    s += v[j];
  }
  unsigned x = s;
#pragma unroll
  for (int d = 1; d < 32; d <<= 1) {
    unsigned y = __shfl_up(x, d, 32);
    if (lane >= (unsigned)d) x += y;
  }
  if (lane == 31u) s_warp[wave] = x;
  __syncthreads();
  if (wave == 0u) {
    unsigned t2 = s_warp[lane];
#pragma unroll
    for (int d = 1; d < 32; d <<= 1) {
      unsigned y = __shfl_up(t2, d, 32);
      if (lane >= (unsigned)d) t2 += y;
    }
    s_warp[lane] = t2;
  }
  __syncthreads();
  unsigned excl = x - s + (wave ? s_warp[wave - 1u] : 0u);
  unsigned cum = excl;
#pragma unroll
  for (int j = 0; j < G; ++j) {
    unsigned nxt = cum + v[j];
    if (cum < k && nxt >= k) {
      *s_selBin = NB - 1u - tid * G - (unsigned)j;
      *s_remain = k - cum;
    }
    cum = nxt;
  }
  __syncthreads();
}

__device__ __forceinline__ void hist_top10(float4 v, float temp, unsigned* hist) {
  float fs[4] = {v.x, v.y, v.z, v.w};
#pragma unroll
  for (int c = 0; c < 4; ++c) {
    unsigned u = fkey(__fdiv_rn(fs[c], temp));
    atomicAdd(&hist[u >> 22], 1u);
  }
}

__global__ void __launch_bounds__(BD, 1)
    topk_sample_kernel(const float* __restrict__ logits, const float* __restrict__ temps,
                       const int* __restrict__ topk, float* __restrict__ out, int V,
                       unsigned halfN) {
  const int row = blockIdx.x;
  const int tid = threadIdx.x;
  const int NV4 = V >> 2;
  const float temp = temps[row];
  unsigned K = (unsigned)topk[0];
  if (K < 1u) K = 1u;
  const float4* rowp = (const float4*)(logits + (size_t)row * (size_t)V);

  __shared__ unsigned s_hist[4096];
  __shared__ unsigned s_warp[32];
  __shared__ unsigned s_selBin;
  __shared__ unsigned s_remain;
  __shared__ float s_bv[32];
  __shared__ int s_bi[32];
#if ASYNC_PIPE
  __shared__ float4 s_buf[2][BD];
#endif

  // ---------------- Pass 1: histogram of top 10 key bits (HBM-bound pass) ----
  s_hist[tid] = 0u;
  __syncthreads();
  {
    const int NT = NV4 / BD;
    const int REM = NV4 - NT * BD;
#if ASYNC_PIPE
    if (NT > 0) async_copy16(rowp + tid, &s_buf[0][tid]);
    for (int t = 0; t < NT; ++t) {
      if (t + 1 < NT) {
        async_copy16(rowp + (size_t)(t + 1) * BD + tid, &s_buf[(t + 1) & 1][tid]);
        WAIT_ASYNC(1);
      } else {
        WAIT_ASYNC(0);
      }
      float4 v = s_buf[t & 1][tid];
      hist_top10(v, temp, s_hist);
    }
#else
    for (int t = 0; t < NT; ++t) {
      float4 v = rowp[(size_t)t * BD + tid];
      hist_top10(v, temp, s_hist);
    }
#endif
    if (tid < REM) {
      float4 v = rowp[(size_t)NT * BD + tid];
      hist_top10(v, temp, s_hist);
    }
  }
  __syncthreads();
  selectStep<1>(s_hist, 1024u, K, s_warp, &s_selBin, &s_remain);
  const unsigned b1 = s_selBin;
  const unsigned r1 = s_remain;

  // ---------------- Pass 2: refine bits [21:12] (L2-resident) ----------------
  s_hist[tid] = 0u;
  __syncthreads();
  for (int i = tid; i < NV4; i += BD) {
    float4 v = rowp[i];
    float fs[4] = {v.x, v.y, v.z, v.w};
#pragma unroll
    for (int c = 0; c < 4; ++c) {
      unsigned u = fkey(__fdiv_rn(fs[c], temp));
      if ((u >> 22) == b1) atomicAdd(&s_hist[(u >> 12) & 1023u], 1u);
    }
  }
  __syncthreads();
  selectStep<1>(s_hist, 1024u, r1, s_warp, &s_selBin, &s_remain);
  const unsigned b2 = s_selBin;
  const unsigned r2 = s_remain;
  const unsigned p2 = (b1 << 10) | b2;

  // ---------------- Pass 3: refine final 12 bits (exact k-th) ----------------
#pragma unroll
  for (int j = 0; j < 4; ++j) s_hist[tid + j * BD] = 0u;
  __syncthreads();
  for (int i = tid; i < NV4; i += BD) {
    float4 v = rowp[i];
    float fs[4] = {v.x, v.y, v.z, v.w};
#pragma unroll
    for (int c = 0; c < 4; ++c) {
      unsigned u = fkey(__fdiv_rn(fs[c], temp));
      if ((u >> 12) == p2) atomicAdd(&s_hist[u & 0xFFFu], 1u);
    }
  }
  __syncthreads();
  selectStep<4>(s_hist, 4096u, r2, s_warp, &s_selBin, &s_remain);
  const unsigned kthU = (p2 << 12) | s_selBin;  // exact bit pattern of kth-largest

  // ---------------- Final pass: Gumbel-argmax over kept elements -------------
  float best = -INFINITY;
  int bidx = INT_MAX;
  for (int i = tid; i < NV4; i += BD) {
    float4 v = rowp[i];
    float fs[4] = {v.x, v.y, v.z, v.w};
#pragma unroll
    for (int c = 0; c < 4; ++c) {
      float sv = __fdiv_rn(fs[c], temp);
      unsigned u = fkey(sv);
      if (u >= kthU) {  // keep iff NOT (scaled < kth)
        int col = (i << 2) + c;
        unsigned flat = (unsigned)row * (unsigned)V + (unsigned)col;
        float sc = sv + gumbel_noise(flat, halfN);
        if (sc > best || (sc == best && col < bidx)) {
          best = sc;
          bidx = col;
        }
      }
    }
  }
  // wave32 reduction, first-index tie-break (matches jnp.argmax)
#pragma unroll
  for (int d = 16; d > 0; d >>= 1) {
    float ov = __shfl_xor(best, d, 32);
    int oi = __shfl_xor(bidx, d, 32);
    if (ov > best || (ov == best && oi < bidx)) {
      best = ov;
      bidx = oi;
    }
  }
  if ((tid & 31) == 0) {
    s_bv[tid >> 5] = best;
    s_bi[tid >> 5] = bidx;
  }
  __syncthreads();
  if (tid < 32) {
    best = s_bv[tid];
    bidx = s_bi[tid];
#pragma unroll
    for (int d = 16; d > 0; d >>= 1) {
      float ov = __shfl_xor(best, d, 32);
      int oi = __shfl_xor(bidx, d, 32);
      if (ov > best || (ov == best && oi < bidx)) {
        best = ov;
        bidx = oi;
      }
    }
    if (tid == 0) out[row] = (float)bidx;
  }
}

extern "C" void kernel_launch(void* const* d_in, const int* in_sizes, int n_in,
                              void* d_out, int out_size, void* d_ws, size_t ws_size,
                              hipStream_t stream) {
  (void)n_in;
  (void)d_ws;
  (void)ws_size;
  const float* logits = (const float*)d_in[0];
  const float* temps = (const float*)d_in[1];
  const int* topk = (const int*)d_in[2];
  float* out = (float*)d_out;

  const int B = in_sizes[1];      // 128 rows (one temperature each)
  const int V = in_sizes[0] / B;  // 128000 vocab
  (void)out_size;
  const unsigned halfN =
      (unsigned)(((unsigned long long)B * (unsigned long long)V) >> 1);

  topk_sample_kernel<<<dim3(B), dim3(BD), 0, stream>>>(logits, temps, topk, out, V, halfN);
}